// TreeSelfAttention_9577777070124
// MI455X (gfx1250) — compile-verified
//
#include <hip/hip_runtime.h>
#include <hip/hip_bf16.h>

// -----------------------------------------------------------------------------
// TreeSelfAttention forward, MI455X (gfx1250, wave32).
//
// Roofline: ~510 MB total traffic (128 MB x-read, 126 MB W-read-once, 256 MB
// broadcast write) vs ~1 GFLOP of GEMM -> purely HBM bound (~22 us at
// 23.3 TB/s). f32 end-to-end is both exact and free; GEMMs use
// V_WMMA_F32_16X16X4_F32 (M=16 == batch fits one tile row exactly).
//
// Key simplification: the LIF spike is data-independent in forward:
//   v = tau*v0 + (s - stop_grad(s)) = tau*v0, so spike = (tau*v0 >= thr).
// s0 is unused in forward.
// -----------------------------------------------------------------------------

typedef __attribute__((ext_vector_type(2))) float v2f;
typedef __attribute__((ext_vector_type(8))) float v8f;

#define TAU   0.95f
#define EPSLN 1e-5f
#define B_DIM 16
#define T_DIM 4096
#define D_DIM 1024
#define K_DIM 2048   // 2*D

// ---------------------------------------------------------------------------
// Kernel 1: leaf means over 256 timesteps, only leaves 0..7, written twice
// (concat([used, used], -1)) into fused3 (B, 8, 2048).
// grid = (4 d-chunks, 8 leaves, 16 batch), block = 256. Fully coalesced.
// ---------------------------------------------------------------------------
__global__ void leaf_mean_kernel(const float* __restrict__ x,
                                 float* __restrict__ fused3) {
  const int d = blockIdx.x * 256 + threadIdx.x;
  const int n = blockIdx.y;
  const int b = blockIdx.z;
  const float* xp = x + ((size_t)(b * T_DIM + n * 256)) * D_DIM + d;
  float acc = 0.0f;
#pragma unroll 8
  for (int t = 0; t < 256; ++t) acc += xp[(size_t)t * D_DIM];
  acc *= (1.0f / 256.0f);
  const size_t o = ((size_t)(b * 8 + n)) * K_DIM + d;
  fused3[o] = acc;           // low half  (k = d)
  fused3[o + D_DIM] = acc;   // high half (k = d + 1024), the concat duplicate
}

// ---------------------------------------------------------------------------
// Kernel 2: one tree level.  proj = fused @ W[node]^T  (M=16,N=1024,K=2048)
// via V_WMMA_F32_16X16X4_F32.  8 waves/block, one 16x16 tile per wave.
// Epilogue: states = (proj + bias) * (tau*v0 >= thr), scattered into next
// level's fused layout: sibling pair (2n,2n+1) -> columns [0..1023|1024..2047].
//
// A-frag (16x4 f32, 2 VGPR): lane L holds A[M=L&15][K = k + (L<16?0:2) + {0,1}]
// B-frag mirrors A with N=lane&15 rows of W (B[k][n] = W[n][k], i.e. W row-major
// already supplies the K-contiguous transposed operand).
// C/D (8 VGPR): c[j] -> row m = j + (L<16?0:8), col n = lane&15.
// ---------------------------------------------------------------------------
__global__ void tree_gemm_kernel(const float* __restrict__ Wn,    // W + start*D*K
                                 const float* __restrict__ bias,  // b + start*D
                                 const float* __restrict__ v0n,   // v0 + start*D
                                 const float* __restrict__ thrn,  // thr + start*D
                                 const float* __restrict__ fusedIn,  // (B,size,K)
                                 float* __restrict__ out,            // next fused or root
                                 int size) {
  const int lane = threadIdx.x & 31;
  const int wave = threadIdx.x >> 5;
  const int node = blockIdx.y;
  const int col0 = blockIdx.x * 128 + wave * 16;
  const int m    = lane & 15;                 // A row (batch) / B column
  const int koff = (lane < 16) ? 0 : 2;       // half-wave K offset
  const int d    = col0 + m;                  // output feature index

  const float* Arow = fusedIn + ((size_t)(m * size + node)) * K_DIM + koff;
  const float* Brow = Wn + ((size_t)node * D_DIM + d) * K_DIM + koff;

  v8f c = {0.f, 0.f, 0.f, 0.f, 0.f, 0.f, 0.f, 0.f};
#pragma unroll 8
  for (int k = 0; k < K_DIM; k += 4) {
    v2f a  = *(const v2f*)(Arow + k);   // 8B-aligned (even element index)
    v2f bb = *(const v2f*)(Brow + k);
    c = __builtin_amdgcn_wmma_f32_16x16x4_f32(
        /*neg_a=*/false, a, /*neg_b=*/false, bb,
        /*c_mod=*/(short)0, c, /*reuse_a=*/false, /*reuse_b=*/false);
  }

  const float bi   = bias[node * D_DIM + d];
  const float mask = (TAU * v0n[node * D_DIM + d] >= thrn[node * D_DIM + d])
                         ? 1.0f : 0.0f;

  const int half = size >> 1;
#pragma unroll
  for (int j = 0; j < 8; ++j) {
    const int bm = j + ((lane < 16) ? 0 : 8);       // batch row of this C elem
    const float s = (c[j] + bi) * mask;
    if (size > 1) {
      const size_t o = ((size_t)(bm * half + (node >> 1))) * K_DIM +
                       (size_t)(node & 1) * D_DIM + d;
      out[o] = s;
    } else {
      out[(size_t)bm * D_DIM + d] = s;              // root (B, D)
    }
  }
}

// ---------------------------------------------------------------------------
// Kernel 3: LayerNorm of root (16 x 1024) -> normed (16 x 1024).
// One block per batch row, 256 threads x float4.
// ---------------------------------------------------------------------------
__global__ void layernorm_kernel(const float* __restrict__ root,
                                 const float* __restrict__ gamma,
                                 const float* __restrict__ beta,
                                 float* __restrict__ normed) {
  __shared__ float red[256];
  const int b = blockIdx.x;
  const int tid = threadIdx.x;

  const float4 v = *(const float4*)(root + (size_t)b * D_DIM + tid * 4);
  red[tid] = v.x + v.y + v.z + v.w;
  __syncthreads();
  for (int s = 128; s > 0; s >>= 1) {
    if (tid < s) red[tid] += red[tid + s];
    __syncthreads();
  }
  const float mean = red[0] * (1.0f / (float)D_DIM);
  __syncthreads();

  const float dx = v.x - mean, dy = v.y - mean, dz = v.z - mean, dw = v.w - mean;
  red[tid] = dx * dx + dy * dy + dz * dz + dw * dw;
  __syncthreads();
  for (int s = 128; s > 0; s >>= 1) {
    if (tid < s) red[tid] += red[tid + s];
    __syncthreads();
  }
  const float var = red[0] * (1.0f / (float)D_DIM);
  const float inv = rsqrtf(var + EPSLN);

  const float4 g  = *(const float4*)(gamma + tid * 4);
  const float4 bt = *(const float4*)(beta + tid * 4);
  float4 o;
  o.x = dx * inv * g.x + bt.x;
  o.y = dy * inv * g.y + bt.y;
  o.z = dz * inv * g.z + bt.z;
  o.w = dw * inv * g.w + bt.w;
  *(float4*)(normed + (size_t)b * D_DIM + tid * 4) = o;
}

// ---------------------------------------------------------------------------
// Kernel 4: broadcast normed[b,:] to out[b,t,:] for all t.  This is the
// bandwidth-dominant step (256 MB of stores); float4 stores, normed rows
// (4 KB each) stay resident in L2.  One block per (b,t) row.
// ---------------------------------------------------------------------------
__global__ void broadcast_kernel(const float* __restrict__ normed,
                                 float* __restrict__ out) {
  const int bt = blockIdx.x;          // b*T + t
  const int b  = bt >> 12;            // T = 4096
  const float4 v =
      *(const float4*)(normed + (size_t)b * D_DIM + threadIdx.x * 4);
  *(float4*)(out + (size_t)bt * D_DIM + threadIdx.x * 4) = v;
}

// ---------------------------------------------------------------------------
extern "C" void kernel_launch(void* const* d_in, const int* in_sizes, int n_in,
                              void* d_out, int out_size, void* d_ws,
                              size_t ws_size, hipStream_t stream) {
  const float* x     = (const float*)d_in[0];
  const float* W     = (const float*)d_in[1];
  const float* bvec  = (const float*)d_in[2];
  const float* thr   = (const float*)d_in[3];
  const float* v0    = (const float*)d_in[4];
  /* d_in[5] = s0: unused in forward (spike depends only on v0, thr) */
  const float* gamma = (const float*)d_in[6];
  const float* beta  = (const float*)d_in[7];
  float* out = (float*)d_out;
  float* ws  = (float*)d_ws;

  // Workspace layout (floats), ~2 MB total.
  float* fused3 = ws;                               // (16, 8, 2048)
  float* fused2 = fused3 + 16 * 8 * K_DIM;          // (16, 4, 2048)
  float* fused1 = fused2 + 16 * 4 * K_DIM;          // (16, 2, 2048)
  float* fused0 = fused1 + 16 * 2 * K_DIM;          // (16, 1, 2048)
  float* root   = fused0 + 16 * 1 * K_DIM;          // (16, 1024)
  float* normed = root + 16 * D_DIM;                // (16, 1024)

  // 1) leaf means (+ concat duplicate)
  leaf_mean_kernel<<<dim3(4, 8, 16), 256, 0, stream>>>(x, fused3);

  // 2) tree levels: (start, size) = (7,8) (3,4) (1,2) (0,1)
  tree_gemm_kernel<<<dim3(8, 8), 256, 0, stream>>>(
      W + (size_t)7 * D_DIM * K_DIM, bvec + 7 * D_DIM, v0 + 7 * D_DIM,
      thr + 7 * D_DIM, fused3, fused2, 8);
  tree_gemm_kernel<<<dim3(8, 4), 256, 0, stream>>>(
      W + (size_t)3 * D_DIM * K_DIM, bvec + 3 * D_DIM, v0 + 3 * D_DIM,
      thr + 3 * D_DIM, fused2, fused1, 4);
  tree_gemm_kernel<<<dim3(8, 2), 256, 0, stream>>>(
      W + (size_t)1 * D_DIM * K_DIM, bvec + 1 * D_DIM, v0 + 1 * D_DIM,
      thr + 1 * D_DIM, fused1, fused0, 2);
  tree_gemm_kernel<<<dim3(8, 1), 256, 0, stream>>>(
      W, bvec, v0, thr, fused0, root, 1);

  // 3) LayerNorm of root
  layernorm_kernel<<<16, 256, 0, stream>>>(root, gamma, beta, normed);

  // 4) broadcast to (B, T, D)
  broadcast_kernel<<<B_DIM * T_DIM, 256, 0, stream>>>(normed, out);
}